// AttentionCombinerConv_81767587381423
// MI455X (gfx1250) — compile-verified
//
#include <hip/hip_runtime.h>
#include <math.h>

// Problem constants (match reference)
#define BATCH   64
#define HW      4096      // 64*64 positions
#define CCH     64        // image channels
#define FEAT    80        // C + BASIS_DIM
#define NHEADS  4
#define CF      64        // conv filters
#define DPH     16        // depth per head
#define NTHREADS 512
#define NWAVES   16
#define TILES_PER_WAVE 16 // 256 tiles of 16 positions / 16 waves

typedef __attribute__((ext_vector_type(2))) float v2f;
typedef __attribute__((ext_vector_type(8))) float v8f;

// Dynamic LDS layout (float offsets)
#define SM_SCORES 0                    // 4096 x 4
#define SM_WQK    (SM_SCORES + HW*4)   // 80 x 16 (cols >=4 zero-padded)
#define SM_WMAT   (SM_WQK + FEAT*16)   // 16 waves x 16 x 16 (cols >=4 zero)
#define SM_ARED   (SM_WMAT + NWAVES*256) // 16 waves x 80 x 4 partial AT
#define SM_QV     (SM_ARED + NWAVES*FEAT*4) // 64 q + 4 const_k (pad to 80)
#define SM_RED    (SM_QV + 80)         // 512 x 4 reduction scratch
#define SM_AT     (SM_RED + NTHREADS*4)// 80 x 4 final AT
#define SM_MS     (SM_AT + FEAT*4)     // max[4], inv_sum[4]
#define SM_TOTAL  (SM_MS + 8)

__global__ __launch_bounds__(NTHREADS)
void attn_fused_kernel(const float* __restrict__ image,
                       const float* __restrict__ state1,
                       const float* __restrict__ state2,
                       const float* __restrict__ extra,
                       const float* __restrict__ Wq,
                       const float* __restrict__ bq,
                       const float* __restrict__ Wk,
                       const float* __restrict__ bk,
                       const float* __restrict__ Wv,
                       const float* __restrict__ bv,
                       const float* __restrict__ basis,
                       float* __restrict__ out)
{
    extern __shared__ float sm[];
    float* s_scores = sm + SM_SCORES;
    float* s_wqk    = sm + SM_WQK;
    float* s_wmat   = sm + SM_WMAT;
    float* s_ared   = sm + SM_ARED;
    float* s_qv     = sm + SM_QV;
    float* s_red    = sm + SM_RED;
    float* s_at     = sm + SM_AT;
    float* s_ms     = sm + SM_MS;

    const int b     = blockIdx.x;
    const int tid   = threadIdx.x;
    const int lane  = tid & 31;       // wave32
    const int wave  = tid >> 5;
    const int ln16  = lane & 15;
    const int khalf = ((lane >> 4) << 1);  // 0 for lanes 0-15, 2 for 16-31

    const float* img_b = image + (size_t)b * HW * CCH;

    // ---------------- Prologue ----------------
    // zero wqk (incl. pad cols) and wmat (pad cols stay zero forever)
    for (int i = tid; i < FEAT * 16; i += NTHREADS)   s_wqk[i]  = 0.f;
    for (int i = tid; i < NWAVES * 256; i += NTHREADS) s_wmat[i] = 0.f;

    // q = concat(state1,state2) @ Wq + bq : 64 outputs, 8 partials each
    {
        const int j    = tid >> 3;
        const int part = tid & 7;
        const float* st1 = state1 + (size_t)b * 512;
        const float* st2 = state2 + (size_t)b * 512;
        float acc = 0.f;
        const int kbeg = part * 128;
        for (int k = kbeg; k < kbeg + 128; ++k) {
            const float sv = (k < 512) ? st1[k] : st2[k - 512];
            acc += sv * Wq[(size_t)k * CF + j];
        }
        s_red[j * 8 + part] = acc;
    }
    __syncthreads();
    if (tid < CF) {
        float s = bq[tid];
        for (int i = 0; i < 8; ++i) s += s_red[tid * 8 + i];
        s_qv[tid] = s;
    }
    __syncthreads();
    // Wqk[c][n] = sum_d Wk[c][n*16+d] * q[n*16+d]; const_k[n] = sum_d bk*q
    if (tid < FEAT * NHEADS) {
        const int c = tid >> 2, n = tid & 3;
        float s = 0.f;
        for (int d = 0; d < DPH; ++d)
            s += Wk[c * CF + n * DPH + d] * s_qv[n * DPH + d];
        s_wqk[c * 16 + n] = s;
    }
    if (tid >= 320 && tid < 320 + NHEADS) {
        const int n = tid - 320;
        float s = 0.f;
        for (int d = 0; d < DPH; ++d) s += bk[n * DPH + d] * s_qv[n * DPH + d];
        s_qv[CF + n] = s;
    }
    __syncthreads();

    // ---------------- Pass 1: scores = feats @ Wqk (WMMA f32 16x16x4) ----
    for (int i = 0; i < TILES_PER_WAVE; ++i) {
        const int p0 = (wave * TILES_PER_WAVE + i) * 16;
        const int p  = p0 + ln16;
        const float* imrow = img_b + (size_t)p * CCH;
        const float* bsrow = basis + (size_t)p * 16;
        // prefetch this wave's next tile (gfx1250 global_prefetch_b8)
        if (i + 1 < TILES_PER_WAVE)
            __builtin_prefetch(img_b + (size_t)(p + 16) * CCH, 0, 1);
        v8f acc = {};
#pragma unroll
        for (int k0 = 0; k0 < FEAT; k0 += 4) {
            const int c = k0 + khalf;   // runtime-uniform image/basis split
            const float* rowp = (c < CCH) ? (imrow + c) : (bsrow + (c - CCH));
            v2f a;  a.x = rowp[0];  a.y = rowp[1];
            v2f bb;
            bb.x = s_wqk[c * 16 + ln16];
            bb.y = s_wqk[(c + 1) * 16 + ln16];
            acc = __builtin_amdgcn_wmma_f32_16x16x4_f32(
                      false, a, false, bb, (short)0, acc, false, false);
        }
        // D layout: VGPR r, lanes0-15 -> (M=r, N=lane); lanes16-31 -> (M=r+8)
        if (ln16 < NHEADS) {
            const int n  = ln16;
            const int mo = (lane >> 4) << 3;
            const float ck = s_qv[CF + n];
#pragma unroll
            for (int r = 0; r < 8; ++r)
                s_scores[(p0 + mo + r) * NHEADS + n] = acc[r] + ck;
        }
    }
    __syncthreads();

    // ---------------- Softmax stats: per-head max, then sum(exp) ----------
    {
        float m0 = -1e30f, m1 = -1e30f, m2 = -1e30f, m3 = -1e30f;
        for (int p = tid; p < HW; p += NTHREADS) {
            m0 = fmaxf(m0, s_scores[p * 4 + 0]);
            m1 = fmaxf(m1, s_scores[p * 4 + 1]);
            m2 = fmaxf(m2, s_scores[p * 4 + 2]);
            m3 = fmaxf(m3, s_scores[p * 4 + 3]);
        }
        s_red[tid*4+0]=m0; s_red[tid*4+1]=m1; s_red[tid*4+2]=m2; s_red[tid*4+3]=m3;
    }
    __syncthreads();
    for (int step = NTHREADS / 2; step > 0; step >>= 1) {
        if (tid < step) {
#pragma unroll
            for (int n = 0; n < 4; ++n)
                s_red[tid*4+n] = fmaxf(s_red[tid*4+n], s_red[(tid+step)*4+n]);
        }
        __syncthreads();
    }
    if (tid < 4) s_ms[tid] = s_red[tid];
    __syncthreads();
    {
        const float x0 = s_ms[0], x1 = s_ms[1], x2 = s_ms[2], x3 = s_ms[3];
        float a0 = 0.f, a1 = 0.f, a2 = 0.f, a3 = 0.f;
        for (int p = tid; p < HW; p += NTHREADS) {
            a0 += __expf(s_scores[p*4+0] - x0);
            a1 += __expf(s_scores[p*4+1] - x1);
            a2 += __expf(s_scores[p*4+2] - x2);
            a3 += __expf(s_scores[p*4+3] - x3);
        }
        __syncthreads();  // s_red max-phase fully consumed before overwrite
        s_red[tid*4+0]=a0; s_red[tid*4+1]=a1; s_red[tid*4+2]=a2; s_red[tid*4+3]=a3;
    }
    __syncthreads();
    for (int step = NTHREADS / 2; step > 0; step >>= 1) {
        if (tid < step) {
#pragma unroll
            for (int n = 0; n < 4; ++n)
                s_red[tid*4+n] += s_red[(tid+step)*4+n];
        }
        __syncthreads();
    }
    if (tid < 4) s_ms[4 + tid] = 1.0f / s_red[tid];
    __syncthreads();

    // ---------------- Pass 2: AT[c][n] = sum_p e[p,n]*feats[p,c] (WMMA) ----
    const float mx0 = s_ms[0], mx1 = s_ms[1], mx2 = s_ms[2], mx3 = s_ms[3];
    const float is0 = s_ms[4], is1 = s_ms[5], is2 = s_ms[6], is3 = s_ms[7];
    float* wm = s_wmat + wave * 256;   // per-wave 16x16 weight tile
    float* mean_out = out + BATCH * 320 + (size_t)b * HW;

    v8f atacc[5];
#pragma unroll
    for (int mt = 0; mt < 5; ++mt) atacc[mt] = (v8f){};

    for (int i = 0; i < TILES_PER_WAVE; ++i) {
        const int p0 = (wave * TILES_PER_WAVE + i) * 16;
        if (lane < 16) {
            const int p = p0 + lane;
            const float e0 = __expf(s_scores[p*4+0] - mx0);
            const float e1 = __expf(s_scores[p*4+1] - mx1);
            const float e2 = __expf(s_scores[p*4+2] - mx2);
            const float e3 = __expf(s_scores[p*4+3] - mx3);
            wm[lane*16+0]=e0; wm[lane*16+1]=e1; wm[lane*16+2]=e2; wm[lane*16+3]=e3;
            mean_out[p] = 0.25f * (e0*is0 + e1*is1 + e2*is2 + e3*is3);
        }
        // AT^T chunk GEMM: M=channel (5 tiles of 16), K=16 positions, N=heads
#pragma unroll
        for (int mt = 0; mt < 5; ++mt) {
            const int c = mt * 16 + ln16;
#pragma unroll
            for (int k0 = 0; k0 < 16; k0 += 4) {
                const int pa = p0 + k0 + khalf;
                v2f a;
                if (mt < 4) {  // compile-time uniform per mt
                    a.x = img_b[(size_t)pa * CCH + c];
                    a.y = img_b[(size_t)(pa + 1) * CCH + c];
                } else {
                    a.x = basis[(size_t)pa * 16 + (c - CCH)];
                    a.y = basis[(size_t)(pa + 1) * 16 + (c - CCH)];
                }
                v2f bb;
                bb.x = wm[(k0 + khalf) * 16 + ln16];
                bb.y = wm[(k0 + khalf + 1) * 16 + ln16];
                atacc[mt] = __builtin_amdgcn_wmma_f32_16x16x4_f32(
                                false, a, false, bb, (short)0, atacc[mt],
                                false, false);
            }
        }
    }
    // dump per-wave AT partials (only head cols valid)
    if (ln16 < NHEADS) {
        float* ap = s_ared + wave * (FEAT * 4);
        const int n  = ln16;
        const int mo = (lane >> 4) << 3;
#pragma unroll
        for (int mt = 0; mt < 5; ++mt)
#pragma unroll
            for (int r = 0; r < 8; ++r)
                ap[(mt * 16 + mo + r) * 4 + n] = atacc[mt][r];
    }
    __syncthreads();
    if (tid < FEAT * NHEADS) {
        float s = 0.f;
        for (int w2 = 0; w2 < NWAVES; ++w2) s += s_ared[w2 * (FEAT*4) + tid];
        s_at[tid] = s;
    }
    __syncthreads();

    // ---------------- Epilogue: pooled + extra -> out[b, 0:320] -----------
    if (tid < CF) {
        const int n = tid >> 4;                 // head of output channel
        const float inv = s_ms[4 + n];
        float s = 0.f;
        for (int c = 0; c < FEAT; ++c) s += s_at[c * 4 + n] * Wv[c * CF + tid];
        out[(size_t)b * 320 + tid] = s * inv + bv[tid];
    } else if (tid < CF + 256) {
        const int i = tid - CF;
        out[(size_t)b * 320 + CF + i] = extra[(size_t)b * 256 + i];
    }
}

extern "C" void kernel_launch(void* const* d_in, const int* in_sizes, int n_in,
                              void* d_out, int out_size, void* d_ws, size_t ws_size,
                              hipStream_t stream) {
    (void)in_sizes; (void)n_in; (void)out_size; (void)d_ws; (void)ws_size;
    const float* image  = (const float*)d_in[0];
    const float* state1 = (const float*)d_in[1];
    const float* state2 = (const float*)d_in[2];
    const float* extra  = (const float*)d_in[3];
    const float* Wq     = (const float*)d_in[4];
    const float* bq     = (const float*)d_in[5];
    const float* Wk     = (const float*)d_in[6];
    const float* bk     = (const float*)d_in[7];
    const float* Wv     = (const float*)d_in[8];
    const float* bv     = (const float*)d_in[9];
    const float* basis  = (const float*)d_in[10];
    float* out = (float*)d_out;

    const size_t smem = (size_t)SM_TOTAL * sizeof(float);  // ~117 KB (<320 KB/WGP)
    hipFuncSetAttribute(reinterpret_cast<const void*>(attn_fused_kernel),
                        hipFuncAttributeMaxDynamicSharedMemorySize, (int)smem);
    attn_fused_kernel<<<BATCH, NTHREADS, smem, stream>>>(
        image, state1, state2, extra, Wq, bq, Wk, bk, Wv, bv, basis, out);
}